// Routing_33225867002121
// MI455X (gfx1250) — compile-verified
//
#include <hip/hip_runtime.h>
#include <math.h>

// Capsule dynamic routing, fully fused recompute strategy for MI455X (gfx1250).
// B=64 batches, Ni=1024 in-caps (Di=16), Nj=32 out-caps (Dj=32), 3 routing iters.
//
// Roofline: recomputing inputs_hat (5 passes over the 64MB W, L2-resident after
// pass 1) costs ~320-400MB of traffic (~17us @ 23.3TB/s) vs ~1.5GB (~65us) if
// hat were materialized in HBM. fp32 WMMA 16x16x4 keeps reference precision;
// 10.7 GFLOP of fp32 matrix work is roughly balanced against that floor.
// W staging uses the gfx1250 async global->LDS path (ASYNCcnt) so the 64KB/i
// copy bypasses VGPRs and overlaps the x staging and softmax of the same iteration.

#define B_     64
#define NI_    1024
#define DI_    16
#define NJ_    32
#define DJ_    32
#define JM_    (NJ_ * DJ_)      // 1024 flattened (j,m) columns
#define CHUNK_ 32               // i's per workgroup
#define BT_    16               // batch tile per workgroup
#define EPS_   1e-7f

typedef float v2f __attribute__((ext_vector_type(2)));
typedef float v8f __attribute__((ext_vector_type(8)));
typedef int b128_t __attribute__((vector_size(16)));   // matches builtin's V4i

// Only probe builtins on the DEVICE pass; the host pass of the HIP dual
// compilation does not expose AMDGPU builtins to __has_builtin.
#if defined(__HIP_DEVICE_COMPILE__)
#if !__has_builtin(__builtin_amdgcn_wmma_f32_16x16x4_f32)
#error "device: gfx1250 toolchain does not expose __builtin_amdgcn_wmma_f32_16x16x4_f32"
#endif
#if __has_builtin(__builtin_amdgcn_global_load_async_to_lds_b128)
#define HAVE_ASYNC_LDS 1
#endif
#if __has_builtin(__builtin_amdgcn_s_wait_asynccnt)
#define HAVE_WAIT_ASYNC 1
#endif
#endif

__device__ __forceinline__ v8f wmma_f32_k4(v2f a, v2f b, v8f c) {
  // D = A(16x4) * B(4x16) + C, fp32, wave32
  return __builtin_amdgcn_wmma_f32_16x16x4_f32(
      /*neg_a=*/false, a, /*neg_b=*/false, b,
      /*c_mod=*/(short)0, c, /*reuse_a=*/false, /*reuse_b=*/false);
}

// Direct global->LDS 128-bit async copy (gfx1250 GLOBAL_LOAD_ASYNC_TO_LDS_B128,
// tracked by ASYNCcnt). Falls back to a VGPR round-trip if unavailable.
// AS1/AS3 pointers are formed via integer casts: for AMDGPU generic pointers,
// the numeric value IS the AS1 address, and flat-LDS addr[31:0] IS the LDS
// offset (aperture rules, ISA 10.2). Pointee type matches the builtin's V4i.
__device__ __forceinline__ void async_copy_b128(const float* gsrc, float* ldst) {
#if defined(HAVE_ASYNC_LDS)
  __builtin_amdgcn_global_load_async_to_lds_b128(
      (__attribute__((address_space(1))) b128_t*)(uintptr_t)gsrc,
      (__attribute__((address_space(3))) b128_t*)(uintptr_t)ldst,
      /*offset=*/0, /*cpol=*/0);
#else
  float4 v = *(const float4*)gsrc;
  *(float4*)ldst = v;
#endif
}

__device__ __forceinline__ void wait_async_copies() {
#if defined(HAVE_ASYNC_LDS)
#if defined(HAVE_WAIT_ASYNC)
  __builtin_amdgcn_s_wait_asynccnt(0);
#else
  asm volatile("s_wait_asynccnt 0" ::: "memory");
#endif
#endif
}

// ---------------------------------------------------------------------------
__global__ void routing_zero_kernel(float* __restrict__ p, int n) {
  int i = blockIdx.x * blockDim.x + threadIdx.x;
  if (i < n) p[i] = 0.0f;
}

// ---------------------------------------------------------------------------
// pass A: outRaw[b,j,m] += sum_i softmax_j(bLog[b,i,:]) * hat[b,i,j,m]
// hat recomputed on the fly with fp32 WMMA. One WG = (16 batches) x (32 i's).
__global__ __launch_bounds__(256) void routing_pass_a(
    const float* __restrict__ X,     // [B, Ni, Di]
    const float* __restrict__ W,     // [Ni, Nj, Di, Dj]
    const float* __restrict__ bLog,  // [B, Ni, Nj]
    float* __restrict__ outRaw)      // [B, Nj, Dj] (atomic accum)
{
  __shared__ __align__(16) float lds_w[DI_ * JM_];   // 64 KB: W[i] as (j,n,m)
  __shared__ __align__(16) float lds_x[BT_ * DI_];   // 1 KB : x[b-tile, i, :]
  __shared__ float lds_c[BT_ * NJ_];                 // 2 KB : softmax coefs

  const int tid  = threadIdx.x;
  const int lane = tid & 31;
  const int wave = tid >> 5;
  const int hi   = lane >> 4;       // 0: lanes 0-15, 1: lanes 16-31
  const int lo   = lane & 15;
  const int b0   = blockIdx.y * BT_;
  const int i0   = blockIdx.x * CHUNK_;

  v8f acc[8];
  const v8f vzero = {};
#pragma unroll
  for (int t = 0; t < 8; ++t) acc[t] = vzero;

  for (int ic = 0; ic < CHUNK_; ++ic) {
    const int i = i0 + ic;
    __syncthreads();  // all lds_w/lds_x/lds_c consumers of prev iter done

    // ---- stage W[i] (16x1024 fp32 = 64KB) into LDS: async global->LDS ----
    const float* wsrc = W + (size_t)i * (DI_ * JM_);
#pragma unroll
    for (int q = 0; q < 16; ++q) {
      int f4 = (tid + q * 256) * 4;                // 4096 float4 total
      async_copy_b128(wsrc + f4, lds_w + f4);
    }

    // ---- stage x[b-tile, i, 0:16] ----
    if (tid < 64) {
      int bb = tid >> 2, n4 = (tid & 3) * 4;
      float4 v = *(const float4*)(X + ((size_t)(b0 + bb) * NI_ + i) * DI_ + n4);
      *(float4*)(lds_x + bb * DI_ + n4) = v;
    }

    // ---- softmax over j for each batch in the tile ----
    if (tid < BT_) {
      const float* bp = bLog + ((size_t)(b0 + tid) * NI_ + i) * NJ_;
      float mx = -1e30f;
#pragma unroll
      for (int j = 0; j < NJ_; ++j) mx = fmaxf(mx, bp[j]);
      float s = 0.0f;
#pragma unroll
      for (int j = 0; j < NJ_; ++j) s += __expf(bp[j] - mx);
      float inv = 1.0f / s;
#pragma unroll
      for (int j = 0; j < NJ_; ++j) lds_c[tid * NJ_ + j] = __expf(bp[j] - mx) * inv;
    }

    wait_async_copies();   // each wave drains its own ASYNCcnt ...
    __syncthreads();       // ... then cross-wave visibility via barrier

    // ---- hat tiles via WMMA; scale by c and accumulate in registers ----
    // wave owns N-tiles [wave*8, wave*8+8); each tile: M=16 batches, N=16 cols
#pragma unroll
    for (int t = 0; t < 8; ++t) {
      const int tileN = wave * 8 + t;
      const int col   = tileN * 16 + lo;           // flattened (j,m)
      const int j     = col >> 5;
      const int m     = col & 31;
      v8f c8 = vzero;
#pragma unroll
      for (int kk = 0; kk < 4; ++kk) {
        const int nb = kk * 4 + 2 * hi;            // K index striping (p + 2*hi)
        v2f a, b;
        a.x = lds_x[lo * DI_ + nb + 0];
        a.y = lds_x[lo * DI_ + nb + 1];
        b.x = lds_w[j * (DI_ * DJ_) + (nb + 0) * DJ_ + m];
        b.y = lds_w[j * (DI_ * DJ_) + (nb + 1) * DJ_ + m];
        c8 = wmma_f32_k4(a, b, c8);
      }
#pragma unroll
      for (int r = 0; r < 8; ++r) {
        const int batch = r + 8 * hi;              // C layout: M = r + 8*hi
        acc[t][r] += c8[r] * lds_c[batch * NJ_ + j];
      }
    }
  }

  // ---- flush register accumulators: one atomicAdd per owned element ----
#pragma unroll
  for (int t = 0; t < 8; ++t) {
    const int tileN = wave * 8 + t;
    const int col   = tileN * 16 + lo;
#pragma unroll
    for (int r = 0; r < 8; ++r) {
      const int batch = b0 + r + 8 * hi;
      atomicAdd(&outRaw[(size_t)batch * JM_ + col], acc[t][r]);
    }
  }
}

// ---------------------------------------------------------------------------
// squash: out[b,j,:] = (||s||^2/(1+||s||^2)/sqrt(||s||^2+eps)) * s
__global__ void routing_squash(const float* __restrict__ raw,
                               float* __restrict__ out) {
  int idx = blockIdx.x * blockDim.x + threadIdx.x;  // one (b,j) per thread
  if (idx >= B_ * NJ_) return;
  const float* s = raw + (size_t)idx * DJ_;
  float sq = 0.0f;
#pragma unroll
  for (int m = 0; m < DJ_; ++m) sq += s[m] * s[m];
  float scale = sq / (1.0f + sq) / sqrtf(sq + EPS_);
#pragma unroll
  for (int m = 0; m < DJ_; ++m) out[(size_t)idx * DJ_ + m] = scale * s[m];
}

// ---------------------------------------------------------------------------
// pass B: bLog[b,i,j] += sum_m v[b,j,m] * hat[b,i,j,m], hat via WMMA again.
__global__ __launch_bounds__(256) void routing_pass_b(
    const float* __restrict__ X,    // [B, Ni, Di]
    const float* __restrict__ W,    // [Ni, Nj, Di, Dj]
    const float* __restrict__ V,    // [B, Nj, Dj] (squashed outputs)
    float* __restrict__ bLog)       // [B, Ni, Nj] (rmw, WG-exclusive slices)
{
  __shared__ __align__(16) float lds_v[BT_ * JM_];   // 64 KB: v for batch tile
  __shared__ __align__(16) float lds_x[BT_ * DI_];   // 1 KB
  __shared__ float lds_agree[BT_ * NJ_];             // 2 KB

  const int tid  = threadIdx.x;
  const int lane = tid & 31;
  const int wave = tid >> 5;
  const int hi   = lane >> 4;
  const int lo   = lane & 15;
  const int b0   = blockIdx.y * BT_;
  const int i0   = blockIdx.x * CHUNK_;

  // ---- stage V for this batch tile once (contiguous 16K floats, async) ----
  const float* vsrc = V + (size_t)b0 * JM_;
#pragma unroll
  for (int q = 0; q < 16; ++q) {
    int f4 = (tid + q * 256) * 4;
    async_copy_b128(vsrc + f4, lds_v + f4);
  }
  wait_async_copies();

  const v8f vzero = {};
  for (int ic = 0; ic < CHUNK_; ++ic) {
    const int i = i0 + ic;
    __syncthreads();  // also covers the initial V staging on ic==0

    lds_agree[tid] = 0.0f;
    lds_agree[tid + 256] = 0.0f;
    if (tid < 64) {
      int bb = tid >> 2, n4 = (tid & 3) * 4;
      float4 v = *(const float4*)(X + ((size_t)(b0 + bb) * NI_ + i) * DI_ + n4);
      *(float4*)(lds_x + bb * DI_ + n4) = v;
    }
    __syncthreads();

    const float* wi = W + (size_t)i * (DI_ * JM_);  // streamed via L2
    if (ic + 1 < CHUNK_) {
      // hint the next i's W block toward the caches (global_prefetch_b8)
      __builtin_prefetch(wi + DI_ * JM_ + tid * 64, 0, 0);
    }
#pragma unroll
    for (int t = 0; t < 8; ++t) {
      const int tileN = wave * 8 + t;
      const int col   = tileN * 16 + lo;
      const int j     = col >> 5;
      const int m     = col & 31;
      v8f c8 = vzero;
#pragma unroll
      for (int kk = 0; kk < 4; ++kk) {
        const int nb = kk * 4 + 2 * hi;
        v2f a, b;
        a.x = lds_x[lo * DI_ + nb + 0];
        a.y = lds_x[lo * DI_ + nb + 1];
        b.x = wi[j * (DI_ * DJ_) + (nb + 0) * DJ_ + m];
        b.y = wi[j * (DI_ * DJ_) + (nb + 1) * DJ_ + m];
        c8 = wmma_f32_k4(a, b, c8);
      }
      // agreement: reduce hat*v over this tile's 16 columns (same j per lane grp)
#pragma unroll
      for (int r = 0; r < 8; ++r) {
        const int batch = r + 8 * hi;
        float val = c8[r] * lds_v[batch * JM_ + col];
        val += __shfl_xor(val, 1, 16);
        val += __shfl_xor(val, 2, 16);
        val += __shfl_xor(val, 4, 16);
        val += __shfl_xor(val, 8, 16);
        if (lo == 0) lds_agree[batch * NJ_ + j] += val;  // wave-exclusive j
      }
    }
    __syncthreads();

    // write back agreement into routing logits (WG-exclusive (b,i,j) slice)
    {
      int idx = tid;                         // entries 0..255
      int bb = idx >> 5, j = idx & 31;
      bLog[((size_t)(b0 + bb) * NI_ + i) * NJ_ + j] += lds_agree[bb * NJ_ + j];
      idx = tid + 256;                       // entries 256..511
      bb = idx >> 5; j = idx & 31;
      bLog[((size_t)(b0 + bb) * NI_ + i) * NJ_ + j] += lds_agree[bb * NJ_ + j];
    }
  }
}

// ---------------------------------------------------------------------------
extern "C" void kernel_launch(void* const* d_in, const int* in_sizes, int n_in,
                              void* d_out, int out_size, void* d_ws, size_t ws_size,
                              hipStream_t stream) {
  const float* X = (const float*)d_in[0];   // [64,1024,16]
  const float* W = (const float*)d_in[1];   // [1024,32,16,32]
  float* out = (float*)d_out;               // [64,32,32]

  // workspace: routing logits (8 MB) + raw pre-squash outputs (256 KB)
  float* bLog = (float*)d_ws;                         // B*Ni*Nj floats
  float* raw  = bLog + (size_t)B_ * NI_ * NJ_;        // B*Nj*Dj floats

  const int nB = B_ * NI_ * NJ_;      // 2,097,152
  const int nO = B_ * NJ_ * DJ_;      // 65,536

  routing_zero_kernel<<<(nB + 255) / 256, 256, 0, stream>>>(bLog, nB);

  dim3 grid(NI_ / CHUNK_, B_ / BT_);  // (32, 4)
  for (int it = 0; it < 3; ++it) {
    routing_zero_kernel<<<(nO + 255) / 256, 256, 0, stream>>>(raw, nO);
    routing_pass_a<<<grid, 256, 0, stream>>>(X, W, bLog, raw);
    routing_squash<<<(B_ * NJ_ + 255) / 256, 256, 0, stream>>>(raw, out);
    if (it < 2)
      routing_pass_b<<<grid, 256, 0, stream>>>(X, W, out, bLog);
  }
}